// TopKSAE_24060406792829
// MI455X (gfx1250) — compile-verified
//
#include <hip/hip_runtime.h>
#include <stdint.h>

#define NT   4096
#define ACT  768
#define DICT 24576
#define TOPK 32
#define EPSV 1e-5f
#define L1C  0.01f

// d_out float offsets (return order: sae_out, acts_topk, 5 scalars, sae_error)
#define SAE_OFF  0
#define ACTS_OFF (NT*ACT)                // 3145728
#define SCAL_OFF (ACTS_OFF + NT*DICT)   // 103809024
#define ERR_OFF  (SCAL_OFF + 5)         // 103809029 (holds xn until final overwrite)

// workspace float offsets
#define WS_MEAN 0
#define WS_STD  (NT)
#define WS_L2   (2*NT)
#define WS_L1   (3*NT)
#define WS_L0   (4*NT)
#define WS_TKV  (5*NT)             // NT*TOPK floats
#define WS_TKI  (5*NT + NT*TOPK)   // NT*TOPK ints (reinterpreted)

typedef float v2f __attribute__((ext_vector_type(2)));
typedef float v8f __attribute__((ext_vector_type(8)));

// ---------------- Kernel 0: per-token mean/std (ddof=1) + xn ----------------
__global__ __launch_bounds__(256) void k_stats(const float* __restrict__ x,
                                               float* __restrict__ out,
                                               float* __restrict__ ws) {
  __shared__ float s1[256], s2[256];
  int row = blockIdx.x, tid = threadIdx.x;
  const float* xr = x + (size_t)row * ACT;
  float a0 = xr[tid], a1 = xr[tid + 256], a2 = xr[tid + 512];
  s1[tid] = a0 + a1 + a2;
  s2[tid] = a0 * a0 + a1 * a1 + a2 * a2;
  __syncthreads();
  for (int st = 128; st > 0; st >>= 1) {
    if (tid < st) { s1[tid] += s1[tid + st]; s2[tid] += s2[tid + st]; }
    __syncthreads();
  }
  float mean = s1[0] / (float)ACT;
  float var  = fmaxf((s2[0] - (float)ACT * mean * mean) / (float)(ACT - 1), 0.f);
  float sd   = sqrtf(var);
  float inv  = 1.f / (sd + EPSV);
  if (tid == 0) { ws[WS_MEAN + row] = mean; ws[WS_STD + row] = sd; }
  float* xn = out + ERR_OFF + (size_t)row * ACT;
  xn[tid]       = (a0 - mean) * inv;
  xn[tid + 256] = (a1 - mean) * inv;
  xn[tid + 512] = (a2 - mean) * inv;
}

// ------- Kernel 1: acts = relu((xn - b_dec) @ W_enc) via f32 WMMA ----------
// Block: 256 threads (8 waves). Block tile 64(M) x 64(N), BK = 16.
// Wave (wm in 0..3, wn in 0..1) computes rows wm*16..+16, cols wn*32..+32.
// B tile is a pure copy -> staged with GLOBAL_LOAD_ASYNC_TO_LDS_B128
// (ASYNCcnt path), overlapped with the A tile's VALU staging (b_dec subtract).
__global__ __launch_bounds__(256) void k_enc(const float* __restrict__ xn,
                                             const float* __restrict__ bdec,
                                             const float* __restrict__ W,
                                             float* __restrict__ acts) {
  __shared__ float A[64][17];
  __shared__ float B[16][64];   // stride 64 keeps 16B-aligned rows for b128 async
  int tid = threadIdx.x;
  int lane = tid & 31, w = tid >> 5;
  int g = lane >> 4, m = lane & 15;
  int wm = w & 3, wn = w >> 2;
  int rowBase = blockIdx.y * 64;
  int colBase = blockIdx.x * 64;

  // Per-thread B staging addresses (constant across K loop except row offset)
  int br = tid >> 4;            // 0..15 (B-tile row)
  int bc = (tid & 15) * 4;      // 0..60 (B-tile col, 4 floats)
  unsigned ldsB = (unsigned)(uintptr_t)&B[br][bc];  // allocation-relative LDS offset

  v8f acc0 = {}, acc1 = {};

  for (int kt = 0; kt < ACT; kt += 16) {
    // --- async stage B tile (16x64): global -> LDS DMA, one b128 per thread ---
    {
      unsigned long long ga =
          (unsigned long long)(uintptr_t)(W + (size_t)(kt + br) * DICT + colBase + bc);
      asm volatile("global_load_async_to_lds_b128 %0, %1, off"
                   :: "v"(ldsB), "v"(ga) : "memory");
    }
    // --- stage A tile (64x16) through VALU: xn - b_dec (overlaps async B) ---
#pragma unroll
    for (int i = 0; i < 4; ++i) {
      int e = tid + 256 * i;
      int r = e >> 4, c = e & 15;
      A[r][c] = xn[(size_t)(rowBase + r) * ACT + kt + c] - bdec[kt + c];
    }
    if (kt + 16 < ACT)  // hint next A K-tile (global_prefetch_b8)
      __builtin_prefetch(&xn[(size_t)(rowBase + (tid >> 4)) * ACT + kt + 16], 0, 0);
    asm volatile("s_wait_asynccnt 0x0" ::: "memory");  // B tile landed in LDS
    __syncthreads();                                   // (+ s_wait_dscnt for A stores)
#pragma unroll
    for (int kk = 0; kk < 16; kk += 4) {
      // A frag (ISA 16x4 layout): lanes 0-15 -> K={kk,kk+1}; lanes 16-31 -> K={kk+2,kk+3}
      v2f a, b0, b1;
      a.x = A[wm * 16 + m][kk + 2 * g];
      a.y = A[wm * 16 + m][kk + 2 * g + 1];
      int n0 = wn * 32;
      b0.x = B[kk + 2 * g][n0 + m];
      b0.y = B[kk + 2 * g + 1][n0 + m];
      b1.x = B[kk + 2 * g][n0 + 16 + m];
      b1.y = B[kk + 2 * g + 1][n0 + 16 + m];
      acc0 = __builtin_amdgcn_wmma_f32_16x16x4_f32(false, a, false, b0, (short)0, acc0, false, false);
      acc1 = __builtin_amdgcn_wmma_f32_16x16x4_f32(false, a, false, b1, (short)0, acc1, false, false);
    }
    __syncthreads();   // all ds_loads retired before next iteration's writes
  }
  // store with ReLU (C layout: VGPR j -> M=j (lanes 0-15) / M=j+8 (lanes 16-31))
#pragma unroll
  for (int j = 0; j < 8; ++j) {
    int r = rowBase + wm * 16 + j + 8 * g;
    int c = colBase + wn * 32 + m;
    float v0 = fmaxf(acc0[j], 0.f) + 0.f;  // +0 canonicalizes -0 for key packing
    float v1 = fmaxf(acc1[j], 0.f) + 0.f;
    acts[(size_t)r * DICT + c]      = v0;
    acts[(size_t)r * DICT + c + 16] = v1;
  }
}

// -------- Kernel 2: exact top-32 per row (iterative argmax), rewrite row ----
__global__ __launch_bounds__(1024) void k_topk(float* __restrict__ acts,
                                               float* __restrict__ ws) {
  __shared__ unsigned long long red[1024];
  int row = blockIdx.x, tid = threadIdx.x;
  float* ar = acts + (size_t)row * DICT;
  float v[24];
#pragma unroll
  for (int i = 0; i < 24; ++i) v[i] = ar[tid + (i << 10)];

  float myval = 0.f; int myidx = 0;
  float l1 = 0.f, l0 = 0.f;
  for (int j = 0; j < TOPK; ++j) {
    unsigned long long best = 0ull;
#pragma unroll
    for (int i = 0; i < 24; ++i) {
      float val = v[i];
      if (val >= 0.f) {  // removed slots are -1
        unsigned e = (unsigned)tid + ((unsigned)i << 10);
        unsigned long long key =
            ((unsigned long long)__float_as_uint(val) << 32) | (0xFFFFFFFFu - e);
        best = key > best ? key : best;  // larger val wins; tie -> lower index
      }
    }
    red[tid] = best;
    __syncthreads();
    for (int st = 512; st > 0; st >>= 1) {
      if (tid < st) { unsigned long long o = red[tid + st]; if (o > red[tid]) red[tid] = o; }
      __syncthreads();
    }
    unsigned long long win = red[0];
    __syncthreads();
    float wv = __uint_as_float((unsigned)(win >> 32));
    unsigned we = 0xFFFFFFFFu - (unsigned)(win & 0xFFFFFFFFu);
    if (tid == j) { myval = wv; myidx = (int)we; }
    if (tid == 0) { l1 += wv; l0 += (wv > 0.f) ? 1.f : 0.f; }
    if ((we & 1023u) == (unsigned)tid) {
      int slot = (int)(we >> 10);
#pragma unroll
      for (int i = 0; i < 24; ++i) if (i == slot) v[i] = -1.f;
    }
  }
  if (tid == 0) { ws[WS_L1 + row] = l1; ws[WS_L0 + row] = l0; }
  if (tid < TOPK) {
    ws[WS_TKV + row * TOPK + tid] = myval;
    ((int*)ws)[WS_TKI + row * TOPK + tid] = myidx;
  }
  // dense scatter: zero entire row, then place the 32 selected values
#pragma unroll
  for (int i = 0; i < 24; ++i) ar[tid + (i << 10)] = 0.f;
  __syncthreads();
  if (tid < TOPK) ar[myidx] = myval;
}

// ---- Kernel 3: sparse decoder + sae_out + sae_error + per-row l2 partial ---
__global__ __launch_bounds__(256) void k_dec(const float* __restrict__ x,
                                             const float* __restrict__ bdec,
                                             const float* __restrict__ Wd,
                                             float* __restrict__ out,
                                             float* __restrict__ ws) {
  __shared__ float sval[TOPK];
  __shared__ int   sidx[TOPK];
  __shared__ float sred[256];
  int row = blockIdx.x, tid = threadIdx.x;
  if (tid < TOPK) {
    sval[tid] = ws[WS_TKV + row * TOPK + tid];
    sidx[tid] = ((const int*)ws)[WS_TKI + row * TOPK + tid];
  }
  __syncthreads();
  float mean = ws[WS_MEAN + row], sd = ws[WS_STD + row];
  float r0 = bdec[tid], r1 = bdec[tid + 256], r2 = bdec[tid + 512];
#pragma unroll 4
  for (int j = 0; j < TOPK; ++j) {
    float vj = sval[j];
    const float* wr = Wd + (size_t)sidx[j] * ACT;
    r0 += vj * wr[tid]; r1 += vj * wr[tid + 256]; r2 += vj * wr[tid + 512];
  }
  float* xn = out + ERR_OFF + (size_t)row * ACT;
  float n0 = xn[tid], n1 = xn[tid + 256], n2 = xn[tid + 512];
  float d0 = r0 - n0, d1 = r1 - n1, d2 = r2 - n2;
  sred[tid] = d0 * d0 + d1 * d1 + d2 * d2;
  __syncthreads();
  for (int st = 128; st > 0; st >>= 1) {
    if (tid < st) sred[tid] += sred[tid + st];
    __syncthreads();
  }
  if (tid == 0) ws[WS_L2 + row] = sred[0];
  const float* xr = x + (size_t)row * ACT;
  float* so = out + SAE_OFF + (size_t)row * ACT;
  float o0 = r0 * sd + mean, o1 = r1 * sd + mean, o2 = r2 * sd + mean;
  so[tid] = o0; so[tid + 256] = o1; so[tid + 512] = o2;
  xn[tid]       = xr[tid]       - o0;   // overwrite xn with sae_error
  xn[tid + 256] = xr[tid + 256] - o1;
  xn[tid + 512] = xr[tid + 512] - o2;
}

// ------------- Kernel 4: deterministic final reduction -> 5 scalars ---------
__global__ __launch_bounds__(1024) void k_final(float* __restrict__ out,
                                                const float* __restrict__ ws) {
  __shared__ float s[1024];
  int tid = threadIdx.x;
  float sums[3];
  const int offs[3] = {WS_L2, WS_L1, WS_L0};
  for (int a = 0; a < 3; ++a) {
    float t = 0.f;
    for (int i = 0; i < 4; ++i) t += ws[offs[a] + tid + (i << 10)];
    s[tid] = t;
    __syncthreads();
    for (int st = 512; st > 0; st >>= 1) {
      if (tid < st) s[tid] += s[tid + st];
      __syncthreads();
    }
    sums[a] = s[0];
    __syncthreads();
  }
  if (tid == 0) {
    float l2  = sums[0] / (float)(NT * ACT);
    float l1n = sums[1] / (float)NT;
    float l0  = sums[2] / (float)NT;
    float* sc = out + SCAL_OFF;
    sc[0] = l2;          // loss
    sc[1] = l2;          // l2_loss
    sc[2] = L1C * l1n;   // l1_loss
    sc[3] = l0;          // l0_norm
    sc[4] = l1n;         // l1_norm
  }
}

extern "C" void kernel_launch(void* const* d_in, const int* in_sizes, int n_in,
                              void* d_out, int out_size, void* d_ws, size_t ws_size,
                              hipStream_t stream) {
  (void)in_sizes; (void)n_in; (void)out_size; (void)ws_size;
  const float* x    = (const float*)d_in[0];
  const float* bdec = (const float*)d_in[1];
  const float* We   = (const float*)d_in[2];
  const float* Wd   = (const float*)d_in[3];
  float* out = (float*)d_out;
  float* ws  = (float*)d_ws;

  k_stats<<<NT, 256, 0, stream>>>(x, out, ws);
  dim3 g1(DICT / 64, NT / 64);
  k_enc<<<g1, 256, 0, stream>>>(out + ERR_OFF, bdec, We, out + ACTS_OFF);
  k_topk<<<NT, 1024, 0, stream>>>(out + ACTS_OFF, ws);
  k_dec<<<NT, 256, 0, stream>>>(x, bdec, Wd, out, ws);
  k_final<<<1, 1024, 0, stream>>>(out, ws);
}